// Dis_87677462381194
// MI455X (gfx1250) — compile-verified
//
#include <hip/hip_runtime.h>
#include <hip/hip_bf16.h>
#include <math.h>

typedef __attribute__((ext_vector_type(2))) float v2f;
typedef __attribute__((ext_vector_type(8))) float v8f;

#define N_NODES 100000
#define N_EDGES 3200000
#define IN_DIM  128
#define HID_DIM 64
#define OUT_DIM 2
#define N_GRAPHS 512

// ---------------------------------------------------------------------------
// 1) xw = x @ W_gcn  via fp32 WMMA 16x16x4.
//    Block = 128 threads (4 waves). Block b covers rows 16b..16b+15,
//    wave w covers cols 16w..16w+15. K unrolled: 32 WMMAs per wave.
// ---------------------------------------------------------------------------
__global__ void __launch_bounds__(128)
gcn_gemm_xw(const float* __restrict__ x, const float* __restrict__ W,
            float* __restrict__ xw) {
    const int wave = threadIdx.x >> 5;
    const int lane = threadIdx.x & 31;
    const int row0 = blockIdx.x << 4;    // 16 rows per block
    const int col0 = wave << 4;          // 16 cols per wave
    const int m  = lane & 15;            // M (A) / N (B,D) index
    const int hi = lane >> 4;            // half-wave select
    const int kb = hi << 1;              // K offset inside a K=4 step

    const float* __restrict__ arow = x + (size_t)(row0 + m) * IN_DIM;

    v8f acc = {};
#pragma unroll
    for (int k0 = 0; k0 < IN_DIM; k0 += 4) {
        v2f a, b;
        // A 16x4 layout: lane m holds A[m][kb], A[m][kb+1] in v0,v1
        a.x = arow[k0 + kb];
        a.y = arow[k0 + kb + 1];
        // B 4x16 layout: lane m holds B[kb][n=m], B[kb+1][n=m] in v0,v1
        b.x = W[(size_t)(k0 + kb) * HID_DIM + col0 + m];
        b.y = W[(size_t)(k0 + kb + 1) * HID_DIM + col0 + m];
        acc = __builtin_amdgcn_wmma_f32_16x16x4_f32(
            /*neg_a=*/false, a, /*neg_b=*/false, b,
            /*c_mod=*/(short)0, acc, /*reuse_a=*/false, /*reuse_b=*/false);
    }
    // D layout: VGPR r -> row (r + 8*hi), col m
#pragma unroll
    for (int r = 0; r < 8; ++r) {
        xw[(size_t)(row0 + r + 8 * hi) * HID_DIM + col0 + m] = acc[r];
    }
}

// ---------------------------------------------------------------------------
// 2) degree with self-loop, then dinv = rsqrt(deg)
// ---------------------------------------------------------------------------
__global__ void init_deg(float* __restrict__ deg, int n) {
    int i = blockIdx.x * blockDim.x + threadIdx.x;
    if (i < n) deg[i] = 1.0f;   // self-loop contributes 1
}

__global__ void count_deg(const int* __restrict__ ei, float* __restrict__ deg,
                          int nE) {
    int i = blockIdx.x * blockDim.x + threadIdx.x;
    if (i < nE) atomicAdd(&deg[ei[nE + i]], 1.0f);  // col = target
}

__global__ void make_dinv(float* __restrict__ deg, int n) {
    int i = blockIdx.x * blockDim.x + threadIdx.x;
    if (i < n) {
        float d = deg[i];
        deg[i] = (d > 0.0f) ? rsqrtf(d) : 0.0f;
    }
}

// ---------------------------------------------------------------------------
// 3) self-loop term: z[i][:] = xw[i][:] * dinv[i]^2   (fully initializes z)
// ---------------------------------------------------------------------------
__global__ void selfloop_init_z(const float* __restrict__ xw,
                                const float* __restrict__ dinv,
                                float* __restrict__ z) {
    int idx = blockIdx.x * blockDim.x + threadIdx.x;   // node*64 + d
    int node = idx >> 6;
    float di = dinv[node];
    z[idx] = xw[idx] * di * di;
}

// ---------------------------------------------------------------------------
// 4) edge scatter: one wave32 per edge, 2 dims per lane, atomic add into z
// ---------------------------------------------------------------------------
__global__ void __launch_bounds__(256)
scatter_edges(const int* __restrict__ ei, const float* __restrict__ xw,
              const float* __restrict__ dinv, float* __restrict__ z, int nE) {
    int warp = (blockIdx.x * blockDim.x + threadIdx.x) >> 5;
    int lane = threadIdx.x & 31;
    if (warp >= nE) return;
    int r = ei[warp];        // source node
    int c = ei[nE + warp];   // target node
    float nrm = dinv[r] * dinv[c];
    const float* __restrict__ src = xw + (size_t)r * HID_DIM;
    float* __restrict__ dst = z + (size_t)c * HID_DIM;
    atomicAdd(&dst[lane],      src[lane]      * nrm);
    atomicAdd(&dst[lane + 32], src[lane + 32] * nrm);
}

// ---------------------------------------------------------------------------
// 5) bias + ReLU + mean-pool accumulation (sums, counts)
// ---------------------------------------------------------------------------
__global__ void pool_accum(const float* __restrict__ z,
                           const float* __restrict__ bgcn,
                           const int* __restrict__ batch,
                           float* __restrict__ sums, float* __restrict__ cnts) {
    int idx = blockIdx.x * blockDim.x + threadIdx.x;   // node*64 + d
    int node = idx >> 6;
    int d = idx & 63;
    float v = z[idx] + bgcn[d];
    v = v > 0.0f ? v : 0.0f;
    int b = batch[node];
    atomicAdd(&sums[b * HID_DIM + d], v);
    if (d == 0) atomicAdd(&cnts[b], 1.0f);
}

// ---------------------------------------------------------------------------
// 6) per-graph MLP: g = sums/cnt; h = relu(g@W1+b1); out = sigmoid(h@W2+b2)
//    one 64-thread block per graph
// ---------------------------------------------------------------------------
__global__ void __launch_bounds__(64)
mlp_head(const float* __restrict__ sums, const float* __restrict__ cnts,
         const float* __restrict__ W1, const float* __restrict__ b1,
         const float* __restrict__ W2, const float* __restrict__ b2,
         float* __restrict__ out) {
    __shared__ float g[HID_DIM];
    __shared__ float h[HID_DIM];
    int gi = blockIdx.x;
    int d = threadIdx.x;
    float c = cnts[gi];
    c = c > 1.0f ? c : 1.0f;
    g[d] = sums[gi * HID_DIM + d] / c;
    __syncthreads();
    float acc = b1[d];
#pragma unroll
    for (int k = 0; k < HID_DIM; ++k) acc += g[k] * W1[k * HID_DIM + d];
    h[d] = acc > 0.0f ? acc : 0.0f;
    __syncthreads();
    if (d < OUT_DIM) {
        float a2 = b2[d];
#pragma unroll
        for (int k = 0; k < HID_DIM; ++k) a2 += h[k] * W2[k * OUT_DIM + d];
        out[gi * OUT_DIM + d] = 1.0f / (1.0f + expf(-a2));
    }
}

// ---------------------------------------------------------------------------
extern "C" void kernel_launch(void* const* d_in, const int* in_sizes, int n_in,
                              void* d_out, int out_size, void* d_ws, size_t ws_size,
                              hipStream_t stream) {
    const float* x     = (const float*)d_in[0];
    const int*   ei    = (const int*)d_in[1];
    const int*   batch = (const int*)d_in[2];
    const float* W_gcn = (const float*)d_in[3];
    const float* b_gcn = (const float*)d_in[4];
    const float* W1    = (const float*)d_in[5];
    const float* b1    = (const float*)d_in[6];
    const float* W2    = (const float*)d_in[7];
    const float* b2    = (const float*)d_in[8];
    float* out = (float*)d_out;

    const int nE = in_sizes[1] / 2;                 // 3,200,000
    const int nN = in_sizes[0] / IN_DIM;            // 100,000

    // workspace layout (floats)
    float* ws   = (float*)d_ws;
    float* xw   = ws;                               // N*64
    float* z    = xw + (size_t)N_NODES * HID_DIM;   // N*64
    float* dinv = z + (size_t)N_NODES * HID_DIM;    // N  (deg, then dinv in-place)
    float* sums = dinv + N_NODES;                   // G*64
    float* cnts = sums + N_GRAPHS * HID_DIM;        // G

    // zero the pooling accumulators (graph-capture safe)
    hipMemsetAsync(sums, 0, (size_t)(N_GRAPHS * HID_DIM + N_GRAPHS) * sizeof(float),
                   stream);

    // degree (with self-loop) -> dinv
    init_deg<<<(nN + 255) / 256, 256, 0, stream>>>(dinv, nN);
    count_deg<<<(nE + 255) / 256, 256, 0, stream>>>(ei, dinv, nE);
    make_dinv<<<(nN + 255) / 256, 256, 0, stream>>>(dinv, nN);

    // dense transform via fp32 WMMA: xw = x @ W_gcn
    gcn_gemm_xw<<<nN / 16, 128, 0, stream>>>(x, W_gcn, xw);

    // z = self-loop term (full init), then scatter-add edge messages
    selfloop_init_z<<<(nN * HID_DIM) / 256, 256, 0, stream>>>(xw, dinv, z);
    scatter_edges<<<(nE + 7) / 8, 256, 0, stream>>>(ei, xw, dinv, z, nE);

    // bias + relu + mean pool
    pool_accum<<<(nN * HID_DIM) / 256, 256, 0, stream>>>(z, b_gcn, batch, sums, cnts);

    // tiny MLP head + sigmoid
    mlp_head<<<N_GRAPHS, 64, 0, stream>>>(sums, cnts, W1, b1, W2, b2, out);
}